// StraightRenderer_8512625181252
// MI455X (gfx1250) — compile-verified
//
#include <hip/hip_runtime.h>
#include <hip/hip_bf16.h>
#include <math.h>

typedef __attribute__((ext_vector_type(2))) float v2f;
typedef __attribute__((ext_vector_type(8))) float v8f;

// Per-segment precomputed constants (padded to 48B for clean B128 loads).
struct __attribute__((aligned(16))) SegC {
    float ex, ey, c1;     // t1 = x*ex + y*ey + c1          (= pv . e)
    float a2, b2, c2;     // t2 = x*a2 + y*b2 + p2 + c2     (= |pv|^2)
    float d2, d2inv, s2;  // |e|^2, 1/(|e|^2+1e-5), (1/(2t))^2
    float pad0, pad1, pad2;
};

// ---------------------------------------------------------------------------
// Kernel 1: build the 224-entry segment-constant table from strokes/thickness.
// S=32 strokes, P=8 points -> 7 segments each, 224 total (= 14 tiles of 16).
// ---------------------------------------------------------------------------
__global__ void seg_precompute(const float* __restrict__ strokes,
                               const float* __restrict__ th,
                               SegC* __restrict__ segs) {
    int j = blockIdx.x * blockDim.x + threadIdx.x;
    if (j >= 224) return;
    int s = j / 7;
    int k = j % 7;
    const float* p0 = strokes + (size_t)(s * 8 + k) * 2;
    float vx = fminf(fmaxf(p0[0], 0.0f), 1.0f) * 512.0f;
    float vy = fminf(fmaxf(p0[1], 0.0f), 1.0f) * 512.0f;
    float wx = fminf(fmaxf(p0[2], 0.0f), 1.0f) * 512.0f;
    float wy = fminf(fmaxf(p0[3], 0.0f), 1.0f) * 512.0f;
    float ex = wx - vx, ey = wy - vy;
    float d2 = ex * ex + ey * ey;
    float t  = fmaxf(th[s] * 2.0f + 0.5f, 0.5f);
    float sc = 1.0f / (2.0f * t);

    SegC c;
    c.ex = ex;  c.ey = ey;  c.c1 = -(vx * ex + vy * ey);
    c.a2 = -2.0f * vx;  c.b2 = -2.0f * vy;  c.c2 = vx * vx + vy * vy;
    c.d2 = d2;  c.d2inv = 1.0f / (d2 + 1e-5f);  c.s2 = sc * sc;
    c.pad0 = c.pad1 = c.pad2 = 0.0f;
    segs[j] = c;
}

// ---------------------------------------------------------------------------
// Kernel 2: WMMA distance-field render.
// Wave handles 16 consecutive flat pixels (same x, y = jb..jb+15).
//   A (16x4 f32):  row m = (x, y_m, p2_m, 1)
//     ISA layout: lanes 0-15 hold M=lane, K=0(v0)/K=1(v1);
//                 lanes 16-31 hold M=lane-16, K=2(v0)/K=3(v1).
//   B (4x16 f32):  col j = (ex, ey, 0, 0) for t1 / (a2, b2, 1, 0) for t2
//     ISA layout (mirror of A): lanes 0-15 = N=lane, K=0(v0)/K=1(v1);
//                               lanes 16-31 = N=lane-16, K=2(v0)/K=3(v1).
//   C (16x16 f32): broadcast c1 / c2 per column (lane = N).
// Epilogue per pair uses per-lane (=per-segment-column) constants, then a
// 16-lane min reduction per half-wave recovers the per-pixel result.
// ---------------------------------------------------------------------------
__global__ __launch_bounds__(256) void render_wmma(const SegC* __restrict__ segs,
                                                   float* __restrict__ out) {
    const int lane = threadIdx.x & 31;
    const int wave = (int)((blockIdx.x * blockDim.x + threadIdx.x) >> 5);
    const int pixBase = wave << 4;        // 16 pixels per wave
    const int xi = pixBase >> 9;          // x index (G=512, 16 | 512)
    const int jb = pixBase & 511;         // y base
    const int m  = lane & 15;
    const bool hi = lane >= 16;

    const float x  = (float)xi;
    const float y  = (float)(jb + m);
    const float p2 = __builtin_fmaf(x, x, y * y);

    v2f A;
    A.x = hi ? p2   : x;   // K=0 (lo) / K=2 (hi)
    A.y = hi ? 1.0f : y;   // K=1 (lo) / K=3 (hi)

    float best2[8];
#pragma unroll
    for (int r = 0; r < 8; ++r) best2[r] = 3.0e38f;

    for (int tile = 0; tile < 14; ++tile) {
        const SegC c = segs[tile * 16 + m];

        v2f B1, B2;
        B1.x = hi ? 0.0f : c.ex;
        B1.y = hi ? 0.0f : c.ey;
        B2.x = hi ? 1.0f : c.a2;   // K=2 row carries the p2 coefficient
        B2.y = hi ? 0.0f : c.b2;

        v8f C1, C2;
#pragma unroll
        for (int r = 0; r < 8; ++r) { C1[r] = c.c1; C2[r] = c.c2; }

        v8f T1 = __builtin_amdgcn_wmma_f32_16x16x4_f32(
            false, A, false, B1, (short)0, C1, false, false);
        v8f T2 = __builtin_amdgcn_wmma_f32_16x16x4_f32(
            false, A, false, B2, (short)0, C2, false, false);

#pragma unroll
        for (int r = 0; r < 8; ++r) {
            float t1    = T1[r];
            float frac  = fminf(fmaxf(t1 * c.d2inv, 0.0f), 1.0f);
            float u     = __builtin_fmaf(frac, c.d2, -2.0f * t1);
            float dist2 = __builtin_fmaf(frac, u, T2[r]);
            dist2 = fmaxf(dist2, 0.0f);                 // guard FP round-off
            best2[r] = fminf(best2[r], dist2 * c.s2);   // sqrt hoisted out
        }
    }

    // Min across the 16 segment columns (lanes) in each half-wave; then the
    // single sqrt per pixel. Lanes 0/16 own pixel rows r / r+8.
#pragma unroll
    for (int r = 0; r < 8; ++r) {
        float v = best2[r];
        v = fminf(v, __shfl_xor(v, 1, 32));
        v = fminf(v, __shfl_xor(v, 2, 32));
        v = fminf(v, __shfl_xor(v, 4, 32));
        v = fminf(v, __shfl_xor(v, 8, 32));
        if (m == 0) {
            int row = r + (hi ? 8 : 0);
            out[xi * 512 + jb + row] = fminf(sqrtf(v), 1.0f);
        }
    }
}

extern "C" void kernel_launch(void* const* d_in, const int* in_sizes, int n_in,
                              void* d_out, int out_size, void* d_ws, size_t ws_size,
                              hipStream_t stream) {
    (void)in_sizes; (void)n_in; (void)out_size; (void)ws_size;
    const float* strokes = (const float*)d_in[0];  // [32,8,2]
    const float* th      = (const float*)d_in[1];  // [32]
    SegC*  segs = (SegC*)d_ws;                      // 224 * 48 B
    float* out  = (float*)d_out;                    // [512,512]

    hipLaunchKernelGGL(seg_precompute, dim3(1), dim3(256), 0, stream,
                       strokes, th, segs);
    // 262144 pixels / 16 per wave = 16384 waves; 8 waves (256 thr) per block.
    hipLaunchKernelGGL(render_wmma, dim3(2048), dim3(256), 0, stream,
                       segs, out);
}